// STDReLu_72842645340568
// MI455X (gfx1250) — compile-verified
//
#include <hip/hip_runtime.h>

// ---------------------------------------------------------------------------
// Fused 10-iteration separable-Gaussian stencil for MI455X (gfx1250).
//   u = relu(o); repeat 10x: u = relu(o - lamn * K * rowconv([a,1,a], 1-2u))
// One workgroup per (b,c) plane; whole plane lives in LDS for all iterations
// => HBM traffic is the minimum (read o once from HBM, write u once; per-iter
// o re-reads hit L2).  The [a,1,a] 1-D convs are done on the matrix pipe as
// 16x16-tile x tridiagonal-matrix products via V_WMMA_F32_16X16X4_F32.
// ---------------------------------------------------------------------------

typedef __attribute__((ext_vector_type(2))) float v2f;
typedef __attribute__((ext_vector_type(8))) float v8f;

namespace {
constexpr int Wd = 256;
constexpr int Hd = 256;
constexpr int Cc = 64;
constexpr int NIT = 10;
constexpr int NTHREADS = 512;            // 16 waves (wave32)
constexpr int NWAVES = 16;               // each wave owns a 16-row band
constexpr int STRIDE = 268;              // LDS row stride (floats); 12*r mod 64
                                         // distinct for r=0..15 -> no bank conflicts
constexpr int UOFF = 4;                  // content col 0 at float 4 (guard cols -2,-1)
constexpr int GROWS = 258;               // plane rows -1..256 (zeroed guard rows)
constexpr int STAGE_OFF = GROWS * STRIDE;        // halo staging area
constexpr int LDS_FLOATS = STAGE_OFF + NWAVES * 2 * 256;
constexpr int LDS_BYTES = LDS_FLOATS * 4;        // 309,344 B <= 320 KB / WGP
}

// tridiagonal conv weight: W(0)=1, W(+/-1)=a, else 0
__device__ __forceinline__ float wfun(int d, float a) {
  return (d == 0) ? 1.0f : (((d == 1) || (d == -1)) ? a : 0.0f);
}

extern "C" __global__ void __launch_bounds__(NTHREADS, 1)
std_relu10(const float* __restrict__ o, const float* __restrict__ nb_sigma,
           const float* __restrict__ lam, float* __restrict__ out) {
  extern __shared__ float lds[];
  const int tid  = threadIdx.x;
  const int lane = tid & 31;
  const int wave = tid >> 5;
  const int lh   = lane >> 4;            // lane half (0: lanes 0-15, 1: 16-31)
  const int lm   = lane & 15;
  const int plane = blockIdx.x;          // b*C + c
  const int ch = plane % Cc;

  const float sigma = nb_sigma[ch];
  const float a    = expf(-1.0f / (2.0f * sigma * sigma));
  const float lamn = lam[0] / (0.2f * 3.14159265358979323846f * sigma * sigma);
  const float j_in = 1.0f + 2.0f * a;    // rowconv of constant 1 (interior col)
  const float j_ed = 1.0f + a;           // rowconv of constant 1 (edge col)

  const float* __restrict__ opl = o   + (size_t)plane * (size_t)(Hd * Wd);
  float*       __restrict__ upl = out + (size_t)plane * (size_t)(Hd * Wd);

  // Zero all of LDS once (guard rows/cols must be 0 for zero-padded conv).
  for (int i = tid; i < LDS_FLOATS; i += NTHREADS) lds[i] = 0.0f;
  __syncthreads();

  const int r0 = wave * 16;              // this wave's band: rows r0..r0+15

  // U(0) = relu(o): each wave fills its own band, float4 vectorized.
  for (int r = 0; r < 16; ++r) {
    const float4* src = (const float4*)(opl + (size_t)(r0 + r) * Wd);
    float4* dst = (float4*)(lds + (r0 + r + 1) * STRIDE + UOFF);
    for (int i = lane; i < Wd / 4; i += 32) {
      float4 v = src[i];
      v.x = fmaxf(v.x, 0.f); v.y = fmaxf(v.y, 0.f);
      v.z = fmaxf(v.z, 0.f); v.w = fmaxf(v.w, 0.f);
      dst[i] = v;
    }
  }

  // --- constant matrices in WMMA register layouts (wave32) ------------------
  // Row pass: P = A_ext(16x20) * Kext(20x16), A_ext = U[rows, 16j-2 .. 16j+17]
  //   Kext[k,c] = W(k-2-c)  (horizontal halo folded into the matmul).
  // B 4x16 slice layout (K-major mirror of documented A layout):
  //   VGPR0: lanes0-15 = row k0, lanes16-31 = row k0+2; VGPR1: rows k0+1,k0+3.
  v2f bK[5];
  #pragma unroll
  for (int s = 0; s < 5; ++s) {
    const int k = 4 * s + 2 * lh;
    bK[s].x = wfun(k     - 2 - lm, a);
    bK[s].y = wfun(k + 1 - 2 - lm, a);
  }
  // Column pass: Q = K2(16x16) * R_tile;  K2[m,k] = W(m-k).
  // A 16x4 slice layout (documented): lanes0-15 K=k0,k0+1; lanes16-31 K=k0+2,k0+3.
  v2f aK[4];
  #pragma unroll
  for (int s = 0; s < 4; ++s) {
    const int k = 4 * s + 2 * lh;
    aK[s].x = wfun(lm - k,       a);
    aK[s].y = wfun(lm - (k + 1), a);
  }

  // Per-lane LDS bases.
  const int abase = (r0 + lm + 1) * STRIDE + (UOFF - 2) + 2 * lh; // row-pass A
  const int dbase = (r0 + 8 * lh + 1) * STRIDE + UOFF + lm;       // C/D layout
  const int bbase = (r0 + 2 * lh + 1) * STRIDE + UOFF + lm;       // col-pass B
  float* const stTop = lds + STAGE_OFF + wave * 512;              // R[r0-1, :]
  float* const stBot = stTop + 256;                               // R[r0+16,:]

  __syncthreads();

  for (int it = 0; it < NIT; ++it) {
    // ---------------- row pass: R = rowconv(1-2U) = Jrow - 2*(U*Kext) ------
    // In-place over own band; 2-column tile overlap handled by A lookahead.
    v2f Ac[5], An[5] = {};
    #pragma unroll
    for (int s = 0; s < 5; ++s) Ac[s] = *(const v2f*)(lds + abase + 4 * s);
    for (int j = 0; j < 16; ++j) {
      if (j < 15) {
        #pragma unroll
        for (int s = 0; s < 5; ++s)
          An[s] = *(const v2f*)(lds + abase + 16 * (j + 1) + 4 * s);
      }
      v8f P = {};
      #pragma unroll
      for (int s = 0; s < 5; ++s)
        P = __builtin_amdgcn_wmma_f32_16x16x4_f32(false, Ac[s], false, bK[s],
                                                  (short)0, P, false, false);
      const int gc = 16 * j + lm;
      const float jr = (gc == 0 || gc == Wd - 1) ? j_ed : j_in;
      float* dp = lds + dbase + 16 * j;
      #pragma unroll
      for (int g = 0; g < 8; ++g) dp[g * STRIDE] = fmaf(P[g], -2.0f, jr);
      #pragma unroll
      for (int s = 0; s < 5; ++s) Ac[s] = An[s];
    }
    __syncthreads();

    // -------- stage vertical halo rows (cross-wave reads, race-protected) --
    {
      const float4* tp = (const float4*)(lds + (r0 +  0) * STRIDE + UOFF);
      const float4* bp = (const float4*)(lds + (r0 + 17) * STRIDE + UOFF);
      for (int i = lane; i < Wd / 4; i += 32) {
        ((float4*)stTop)[i] = tp[i];
        ((float4*)stBot)[i] = bp[i];
      }
    }
    __syncthreads();

    // ------------- column pass + pointwise: u = relu(o - lamn*(K2*R)) ------
    const bool last = (it == NIT - 1);
    for (int j = 0; j < 16; ++j) {
      v8f Q = {};
      #pragma unroll
      for (int s = 0; s < 4; ++s) {
        v2f bv;                                  // B slice: R rows 4s..4s+3
        bv.x = lds[bbase + (4 * s)     * STRIDE + 16 * j];
        bv.y = lds[bbase + (4 * s + 1) * STRIDE + 16 * j];
        Q = __builtin_amdgcn_wmma_f32_16x16x4_f32(false, aK[s], false, bv,
                                                  (short)0, Q, false, false);
      }
      // rank-1 vertical halo fixups: D row 0 += a*R[r0-1], row 15 += a*R[r0+16]
      const float ft = stTop[16 * j + lm];
      const float fb = stBot[16 * j + lm];
      Q[0] += (lh == 0) ? a * ft : 0.0f;
      Q[7] += (lh == 1) ? a * fb : 0.0f;

      const float* og = opl + (size_t)(r0 + 8 * lh) * Wd + 16 * j + lm;
      float* dl = lds + dbase + 16 * j;
      float* dg = upl + (size_t)(r0 + 8 * lh) * Wd + 16 * j + lm;
      #pragma unroll
      for (int g = 0; g < 8; ++g) {
        const float u = fmaxf(fmaf(-lamn, Q[g], og[(size_t)g * Wd]), 0.0f);
        if (last) dg[(size_t)g * Wd] = u;       // final iter -> HBM directly
        else      dl[g * STRIDE] = u;           // else back to LDS plane
      }
    }
    // No end-of-iteration barrier needed: next row pass touches only own band;
    // all cross-wave reads are behind the two barriers above.
  }
}

extern "C" void kernel_launch(void* const* d_in, const int* in_sizes, int n_in,
                              void* d_out, int out_size, void* d_ws, size_t ws_size,
                              hipStream_t stream) {
  const float* o    = (const float*)d_in[0];
  const float* sig  = (const float*)d_in[1];
  const float* lamp = (const float*)d_in[2];
  float* out = (float*)d_out;
  const int planes = in_sizes[0] / (Hd * Wd);   // B*C = 1024

  // 302 KB dynamic LDS > default cap; raise it (idempotent, capture-safe).
  hipFuncSetAttribute(reinterpret_cast<const void*>(&std_relu10),
                      hipFuncAttributeMaxDynamicSharedMemorySize, LDS_BYTES);

  std_relu10<<<planes, NTHREADS, LDS_BYTES, stream>>>(o, sig, lamp, out);
}